// AdaptiveGeometricLoss_77189152244308
// MI455X (gfx1250) — compile-verified
//
#include <hip/hip_runtime.h>
#include <math.h>

// ---------------------------------------------------------------------------
// Problem constants (fixed by the reference: B=16, H=W=512, f32)
// ---------------------------------------------------------------------------
#define BATCH 16
#define HH    512
#define WW    512
#define HW    (HH * WW)        // 262144
#define NPIX  (BATCH * HW)     // 4194304

// Workspace byte offsets
#define OFF_SCAL  0                        // float[4]: gmin,gmax,dmin,dmax
#define OFF_TOT   128                      // int[16]:  fg pixel count per sample
#define OFF_MAXC  192                      // int[16]:  max component size per sample
#define OFF_SIMP  256                      // float[4096]: per-block sq-diff partial sums
#define OFF_AREAP (OFF_SIMP  + 4096 * 4)   // float[4096]: per-block area partial sums
#define OFF_LAB   (OFF_AREAP + 4096 * 4)   // int[NPIX]: union-find labels
#define OFF_CNT   (OFF_LAB   + NPIX * 4)   // int[NPIX]: component size histogram

typedef __attribute__((ext_vector_type(2))) float v2f;
typedef __attribute__((ext_vector_type(8))) float v8f;

// Pointer types expected by __builtin_amdgcn_global_load_async_to_lds_b128:
// (int4 addrspace(1)*, int4 addrspace(3)*, imm offset, imm cpol)
typedef int v4i_gcc __attribute__((vector_size(16)));
typedef __attribute__((address_space(1))) v4i_gcc* gbl_v4i_p;
typedef __attribute__((address_space(3))) v4i_gcc* lds_v4i_p;

// LDS tile: 4 DEM rows (center 2 + halo), padded so the 512-float data region
// starts at a 16-byte boundary (col 4) and the row pitch (520*4=2080) is a
// multiple of 16 -> b128 async LDS writes are naturally aligned.
#define TW 520

#if defined(__gfx1250__)
#if __has_builtin(__builtin_amdgcn_global_load_async_to_lds_b128)
#define HAVE_ASYNC_LDS 1
#endif
#endif
#ifndef HAVE_ASYNC_LDS
#define HAVE_ASYNC_LDS 0
#endif

// ---------------------------------------------------------------------------
// Exact f32 wave(32)-wide sum using V_WMMA_F32_16X16X4_F32.
// A is 16x4: lanes 0-15 supply A[m,0]=v_m, lanes 16-31 supply A[m,2]=v_{m+16};
// B = all-ones 4x16 (layout agnostic). Then D[m,n] = v_m + v_{m+16} for all n.
// Per-lane sum of the 8 accumulator VGPRs + shfl_xor(16) -> full 32-lane sum.
// Requires EXEC == all ones (call sites are convergent full blocks).
// ---------------------------------------------------------------------------
__device__ __forceinline__ float wave_sum_wmma(float v) {
    v2f a;  a[0] = v;    a[1] = 0.0f;
    v2f bm; bm[0] = 1.0f; bm[1] = 1.0f;
    v8f c = {0.f, 0.f, 0.f, 0.f, 0.f, 0.f, 0.f, 0.f};
    c = __builtin_amdgcn_wmma_f32_16x16x4_f32(false, a, false, bm,
                                              (short)0, c, false, false);
    float s = c[0] + c[1] + c[2] + c[3] + c[4] + c[5] + c[6] + c[7];
    return s + __shfl_xor(s, 16, 32);
}

// Deterministic float atomic min/max (CAS on bit pattern, compare in float).
__device__ __forceinline__ void atomicMinF(float* addr, float val) {
    unsigned* ua = (unsigned*)addr;
    unsigned old = *ua;
    while (__uint_as_float(old) > val) {
        unsigned assumed = old;
        old = atomicCAS(ua, assumed, __float_as_uint(val));
        if (old == assumed) break;
    }
}
__device__ __forceinline__ void atomicMaxF(float* addr, float val) {
    unsigned* ua = (unsigned*)addr;
    unsigned old = *ua;
    while (__uint_as_float(old) < val) {
        unsigned assumed = old;
        old = atomicCAS(ua, assumed, __float_as_uint(val));
        if (old == assumed) break;
    }
}

// ---------------------------------------------------------------------------
// Stage a 4-row x 512-col DEM strip (rows r0-1 .. r0+2 of sample b) into LDS
// via GLOBAL_LOAD_ASYNC_TO_LDS_B128 (ASYNCcnt path). Out-of-image rows and the
// two halo columns are zero-filled (= lax.conv zero SAME padding).
// 4 rows * 512 floats = 512 x 16B segments; 2 async ops per thread.
// ---------------------------------------------------------------------------
__device__ __forceinline__ void load_dem_tile(const float* __restrict__ dem,
                                              int b, int r0,
                                              float (&tile)[4][TW]) {
    const float* dbase = dem + ((size_t)b << 18);
    #pragma unroll
    for (int k = 0; k < 2; ++k) {
        int seg = threadIdx.x + (k << 8);          // 0..511
        int row = seg >> 7;                        // 0..3
        int xs  = (seg & 127) << 2;                // 0..508 (16B segment)
        int yg  = r0 - 1 + row;
        float* l = &tile[row][4 + xs];
        if ((unsigned)yg < (unsigned)HH) {
            float* g = const_cast<float*>(dbase + (yg << 9) + xs);
#if HAVE_ASYNC_LDS
            __builtin_amdgcn_global_load_async_to_lds_b128(
                (gbl_v4i_p)g, (lds_v4i_p)l, /*offset=*/0, /*cpol=*/0);
#else
            *(float4*)l = *(const float4*)g;
#endif
        } else {
            l[0] = 0.f; l[1] = 0.f; l[2] = 0.f; l[3] = 0.f;
        }
    }
    if (threadIdx.x < 4) {                         // halo columns x=-1, x=512
        tile[threadIdx.x][3]      = 0.f;
        tile[threadIdx.x][4 + WW] = 0.f;
    }
#if HAVE_ASYNC_LDS
#if __has_builtin(__builtin_amdgcn_s_wait_asynccnt)
    __builtin_amdgcn_s_wait_asynccnt(0);
#else
    asm volatile("s_wait_asynccnt 0" ::: "memory");
#endif
#endif
    __syncthreads();
}

// 3x3 stencil taps out of the staged tile: pixel (yloc in {0,1}, x in [0,512)).
#define TAP(dy, dx) tile[1 + yloc + (dy)][4 + x + (dx)]

// ---------------------------------------------------------------------------
// K0: initialize workspace (ws is poisoned to 0xAA by the harness).
// ---------------------------------------------------------------------------
__global__ void k_init(char* ws) {
    int i = blockIdx.x * blockDim.x + threadIdx.x;
    int* counts = (int*)(ws + OFF_CNT);
    for (int idx = i; idx < NPIX; idx += gridDim.x * blockDim.x) counts[idx] = 0;
    if (i < 16) {
        ((int*)(ws + OFF_TOT))[i]  = 0;
        ((int*)(ws + OFF_MAXC))[i] = 0;
    }
    if (i == 0) {
        float* s = (float*)(ws + OFF_SCAL);
        s[0] =  INFINITY;  // gmag min
        s[1] = -INFINITY;  // gmag max
        s[2] =  INFINITY;  // dem min
        s[3] = -INFINITY;  // dem max
    }
}

// ---------------------------------------------------------------------------
// K1: global min/max of gradient magnitude and DEM height.
// Block -> (sample b, 2-row strip). DEM staged through LDS (async path).
// ---------------------------------------------------------------------------
__global__ void k_minmax(const float* __restrict__ dem, char* ws) {
    __shared__ float tile[4][TW];
    const int b  = blockIdx.x >> 8;
    const int r0 = (blockIdx.x & 255) * 2;
    load_dem_tile(dem, b, r0, tile);

    float gmin =  INFINITY, gmax = -INFINITY;
    float dmin =  INFINITY, dmax = -INFINITY;

    #pragma unroll
    for (int k = 0; k < 4; ++k) {
        int idx  = (k << 8) + threadIdx.x;    // 0..1023 over the 2-row strip
        int yloc = idx >> 9;                  // 0..1
        int x    = idx & 511;
        float d00 = TAP(-1,-1), d01 = TAP(-1,0), d02 = TAP(-1,1);
        float d10 = TAP( 0,-1), dc  = TAP( 0,0), d12 = TAP( 0,1);
        float d20 = TAP( 1,-1), d21 = TAP( 1,0), d22 = TAP( 1,1);
        float gx = (d02 - d00) + 2.0f * (d12 - d10) + (d22 - d20);
        float gy = (d20 - d00) + 2.0f * (d21 - d01) + (d22 - d02);
        float g  = sqrtf(gx * gx + gy * gy + 1e-8f);
        gmin = fminf(gmin, g);  gmax = fmaxf(gmax, g);
        dmin = fminf(dmin, dc); dmax = fmaxf(dmax, dc);
    }
    #pragma unroll
    for (int m = 16; m >= 1; m >>= 1) {
        gmin = fminf(gmin, __shfl_xor(gmin, m, 32));
        gmax = fmaxf(gmax, __shfl_xor(gmax, m, 32));
        dmin = fminf(dmin, __shfl_xor(dmin, m, 32));
        dmax = fmaxf(dmax, __shfl_xor(dmax, m, 32));
    }
    __shared__ float red[32];
    int lane = threadIdx.x & 31, wid = threadIdx.x >> 5;
    if (lane == 0) {
        red[wid] = gmin; red[8 + wid] = gmax; red[16 + wid] = dmin; red[24 + wid] = dmax;
    }
    __syncthreads();
    if (threadIdx.x == 0) {
        float a =  INFINITY, bmx = -INFINITY, c =  INFINITY, dmx = -INFINITY;
        #pragma unroll
        for (int w = 0; w < 8; ++w) {
            a = fminf(a, red[w]);      bmx = fmaxf(bmx, red[8 + w]);
            c = fminf(c, red[16 + w]); dmx = fmaxf(dmx, red[24 + w]);
        }
        float* s = (float*)(ws + OFF_SCAL);
        atomicMinF(&s[0], a); atomicMaxF(&s[1], bmx);
        atomicMinF(&s[2], c); atomicMaxF(&s[3], dmx);
    }
}

// ---------------------------------------------------------------------------
// K2: similarity partial sums + per-sample area partial sums.
// DEM staged through LDS; wave reduction via WMMA.
// ---------------------------------------------------------------------------
__global__ void k_feat(const float* __restrict__ pred, const float* __restrict__ dem,
                       char* ws) {
    __shared__ float tile[4][TW];
    const int b  = blockIdx.x >> 8;
    const int r0 = (blockIdx.x & 255) * 2;
    const float* s  = (const float*)(ws + OFF_SCAL);
    const float gmin = s[0], gmax = s[1], dmin = s[2], dmax = s[3];
    const float ginv = 1.0f / (gmax - gmin + 1e-8f);
    const float dinv = 1.0f / (dmax - dmin + 1e-8f);

    load_dem_tile(dem, b, r0, tile);

    const float* pbase = pred + ((size_t)b << 18) + ((size_t)r0 << 9);
    float lsum = 0.0f, larea = 0.0f;
    #pragma unroll
    for (int k = 0; k < 4; ++k) {
        int idx  = (k << 8) + threadIdx.x;
        int yloc = idx >> 9;
        int x    = idx & 511;
        float d00 = TAP(-1,-1), d01 = TAP(-1,0), d02 = TAP(-1,1);
        float d10 = TAP( 0,-1), dc  = TAP( 0,0), d12 = TAP( 0,1);
        float d20 = TAP( 1,-1), d21 = TAP( 1,0), d22 = TAP( 1,1);
        float gx  = (d02 - d00) + 2.0f * (d12 - d10) + (d22 - d20);
        float gy  = (d20 - d00) + 2.0f * (d21 - d01) + (d22 - d02);
        float lap = d01 + d10 + d12 + d21 - 4.0f * dc;
        float gn  = (sqrtf(gx * gx + gy * gy + 1e-8f) - gmin) * ginv;
        float hn  = (dc - dmin) * dinv;
        float cv  = tanhf(lap * 0.1f);
        float p   = pbase[idx];
        float e0 = p - gn, e1 = p - hn, e2 = p - cv;
        lsum  += e0 * e0 + e1 * e1 + e2 * e2;
        larea += p;
    }
    float wsum  = wave_sum_wmma(lsum);    // v_wmma_f32_16x16x4_f32
    float warea = wave_sum_wmma(larea);

    __shared__ float red[16];
    int lane = threadIdx.x & 31, wid = threadIdx.x >> 5;
    if (lane == 0) { red[wid] = wsum; red[8 + wid] = warea; }
    __syncthreads();
    if (threadIdx.x == 0) {
        float a = 0.0f, c = 0.0f;
        #pragma unroll
        for (int w = 0; w < 8; ++w) { a += red[w]; c += red[8 + w]; }
        ((float*)(ws + OFF_SIMP))[blockIdx.x]  = a;
        ((float*)(ws + OFF_AREAP))[blockIdx.x] = c;
    }
}

// ---------------------------------------------------------------------------
// Connectivity: union-find CCL == converged min-label propagation
// (final root of each component is its minimum pixel index, exactly matching
// the reference's iterated min-propagation fixed point).
// ---------------------------------------------------------------------------
__global__ void k_lab_init(char* ws) {
    int i = blockIdx.x * blockDim.x + threadIdx.x;
    if (i < NPIX) ((int*)(ws + OFF_LAB))[i] = i;
}

__device__ __forceinline__ void uf_merge(int* L, int a, int b) {
    while (a != b) {
        if (a < b) { int t = a; a = b; b = t; }   // a > b
        int old = atomicMin(&L[a], b);
        if (old == a) return;                      // linked a -> b
        a = old;                                   // keep merging old's set with b
    }
}

__global__ void k_merge(const float* __restrict__ pred, char* ws) {
    int i = blockIdx.x * blockDim.x + threadIdx.x;
    if (i >= NPIX) return;
    if (!(pred[i] > 0.5f)) return;
    int* L = (int*)(ws + OFF_LAB);
    int local = i & (HW - 1);
    int y = local >> 9, x = local & 511;
    if (x + 1 < WW && pred[i + 1]  > 0.5f) uf_merge(L, i, i + 1);
    if (y + 1 < HH && pred[i + WW] > 0.5f) uf_merge(L, i, i + WW);
}

__global__ void k_flatten(char* ws) {
    int i = blockIdx.x * blockDim.x + threadIdx.x;
    if (i >= NPIX) return;
    int* L = (int*)(ws + OFF_LAB);
    int r = L[i];
    int p = L[r];
    while (p != r) { r = p; p = L[r]; }
    L[i] = r;
}

__global__ void k_count(const float* __restrict__ pred, char* ws) {
    int i = blockIdx.x * blockDim.x + threadIdx.x;
    if (i >= NPIX) return;
    if (!(pred[i] > 0.5f)) return;
    int* L      = (int*)(ws + OFF_LAB);
    int* counts = (int*)(ws + OFF_CNT);
    int* tot    = (int*)(ws + OFF_TOT);
    atomicAdd(&counts[L[i]], 1);
    atomicAdd(&tot[i >> 18], 1);
}

__global__ void k_maxcomp(char* ws) {
    int i = blockIdx.x * blockDim.x + threadIdx.x;
    if (i >= NPIX) return;
    int c = ((int*)(ws + OFF_CNT))[i];
    if (c > 0) atomicMax(&((int*)(ws + OFF_MAXC))[i >> 18], c);
}

// ---------------------------------------------------------------------------
// K8: final combine (single block, WMMA wave reduce for similarity sum).
// ---------------------------------------------------------------------------
__global__ void k_final(char* ws, float* out) {
    const float* simP  = (const float*)(ws + OFF_SIMP);
    const float* areaP = (const float*)(ws + OFF_AREAP);
    const int*   tot   = (const int*)(ws + OFF_TOT);
    const int*   maxc  = (const int*)(ws + OFF_MAXC);

    float v = 0.0f;
    #pragma unroll
    for (int k = 0; k < 16; ++k) v += simP[threadIdx.x + 256 * k];
    float wsum = wave_sum_wmma(v);

    __shared__ float sred[8];
    __shared__ float areas[16];
    int lane = threadIdx.x & 31, wid = threadIdx.x >> 5;
    if (lane == 0) sred[wid] = wsum;
    __syncthreads();
    if (threadIdx.x < 16) {
        float a = 0.0f;
        for (int j = 0; j < 256; ++j) a += areaP[threadIdx.x * 256 + j];
        areas[threadIdx.x] = a;
    }
    __syncthreads();
    if (threadIdx.x == 0) {
        float simsum = 0.0f;
        #pragma unroll
        for (int w = 0; w < 8; ++w) simsum += sred[w];
        float similarity = simsum / (float)(BATCH * 3 * HW);

        float conn = 0.0f;
        for (int b = 0; b < BATCH; ++b) {
            float t = (float)tot[b];
            if (t > 0.0f) conn += 1.0f - (float)maxc[b] / fmaxf(t, 1.0f);
        }
        conn /= (float)BATCH;

        const float tp = (float)HW, tmin = 0.1f * tp, tmax = 0.3f * tp;
        float scale = 0.0f;
        for (int b = 0; b < BATCH; ++b) {
            float a = areas[b];
            scale += fmaxf(a - tmax, 0.0f) + fmaxf(tmin - a, 0.0f);
        }
        scale = (scale / (float)BATCH) / tp;

        out[0] = 0.1f * (similarity + 0.1f * conn + 0.05f * scale);
    }
}

// ---------------------------------------------------------------------------
// Host launcher (graph-capture safe: stream-ordered kernels only)
// ---------------------------------------------------------------------------
extern "C" void kernel_launch(void* const* d_in, const int* in_sizes, int n_in,
                              void* d_out, int out_size, void* d_ws, size_t ws_size,
                              hipStream_t stream) {
    const float* pred = (const float*)d_in[0];   // [16,1,512,512] f32
    const float* dem  = (const float*)d_in[1];   // [16,1,512,512] f32
    float* out = (float*)d_out;
    char* ws   = (char*)d_ws;

    const int T = 256;
    const int gridPix = NPIX / T;   // 16384

    k_init    <<<gridPix, T, 0, stream>>>(ws);
    k_minmax  <<<4096,    T, 0, stream>>>(dem, ws);
    k_feat    <<<4096,    T, 0, stream>>>(pred, dem, ws);
    k_lab_init<<<gridPix, T, 0, stream>>>(ws);
    k_merge   <<<gridPix, T, 0, stream>>>(pred, ws);
    k_flatten <<<gridPix, T, 0, stream>>>(ws);
    k_count   <<<gridPix, T, 0, stream>>>(pred, ws);
    k_maxcomp <<<gridPix, T, 0, stream>>>(ws);
    k_final   <<<1,       T, 0, stream>>>(ws, out);
}